// BeeSender_65687229826041
// MI455X (gfx1250) — compile-verified
//
#include <hip/hip_runtime.h>

typedef __attribute__((ext_vector_type(2))) float v2f;
typedef __attribute__((ext_vector_type(8))) float v8f;

#define NNODES 40000
#define NEDGES 640000
#define NREL   4
#define FDIM   128
#define EMB    128
#define HID    256
#define VOCAB  16
#define BQ     4096

// ---------------------------------------------------------------- utilities
__global__ void zero_kernel(float* __restrict__ p, int n) {
    int i = blockIdx.x * blockDim.x + threadIdx.x;
    if (i < n) p[i] = 0.0f;
}

// per-(relation,dst) in-degree counts (exact in fp32 for degrees << 2^24)
__global__ void count_kernel(const int* __restrict__ edst,
                             const int* __restrict__ etype,
                             float* __restrict__ cnt, int E, int N) {
    int e = blockIdx.x * blockDim.x + threadIdx.x;
    if (e < E) {
        int r = etype[e];
        int d = edst[e];
        __hip_atomic_fetch_add(&cnt[(size_t)r * N + d], 1.0f,
                               __ATOMIC_RELAXED, __HIP_MEMORY_SCOPE_AGENT);
    }
}

// ------------------------------------------------- fp32 WMMA GEMM (full precision)
// C[M,Nc] = A[M,K] @ W[K,Nc] (+bias) (optional relu)
// grid: (M/16, Nc/(16*GWAVES)), block: 32*GWAVES threads (wave32)
#define GWAVES 4
__global__ void wmma_gemm_kernel(const float* __restrict__ A,
                                 const float* __restrict__ W,
                                 const float* __restrict__ bias,
                                 float* __restrict__ C,
                                 int M, int K, int Nc, int do_relu) {
    const int lane = threadIdx.x & 31;
    const int wave = threadIdx.x >> 5;
    const int m0 = blockIdx.x * 16;
    const int n0 = (blockIdx.y * GWAVES + wave) * 16;
    const int lm = lane & 15;                 // row within A tile / col within C tile
    const int kb = (lane >> 4) << 1;          // 0 or 2: K sub-offset per half-wave

    const float* __restrict__ arow = A + (size_t)(m0 + lm) * K + kb;
    const float* __restrict__ wcol = W + (size_t)kb * Nc + (n0 + lm);

    v8f acc = {};
    #pragma unroll 4
    for (int kk = 0; kk < K; kk += 4) {
        // A 16x4 fragment: lane holds A[m0+lm][kk+kb], A[m0+lm][kk+kb+1]
        v2f a = *(const v2f*)(arow + kk);
        // B 4x16 fragment: lane holds W[kk+kb][n0+lm], W[kk+kb+1][n0+lm]
        v2f b;
        b.x = wcol[(size_t)kk * Nc];
        b.y = wcol[(size_t)(kk + 1) * Nc];
        acc = __builtin_amdgcn_wmma_f32_16x16x4_f32(false, a, false, b,
                                                    (short)0, acc, false, false);
    }
    float bv = bias ? bias[n0 + lm] : 0.0f;
    const int mh = (lane >> 4) << 3;          // +8 rows for upper half-wave
    #pragma unroll
    for (int i = 0; i < 8; ++i) {
        float v = acc[i] + bv;
        if (do_relu) v = fmaxf(v, 0.0f);
        C[(size_t)(m0 + i + mh) * Nc + (n0 + lm)] = v;
    }
}

// ---------------------------------------- edge scatter: h[dst] += xr[src]/cnt
// one wave per edge; 4 x 32-lane fp32 atomics cover 128 features
__global__ void scatter_kernel(const int* __restrict__ esrc,
                               const int* __restrict__ edst,
                               const int* __restrict__ etype,
                               const float* __restrict__ cnt,
                               const float* __restrict__ xr,
                               float* __restrict__ hout,
                               int E, int N, int rel) {
    const int lane = threadIdx.x & 31;
    const int wid  = blockIdx.x * (blockDim.x >> 5) + (threadIdx.x >> 5);
    const int nw   = gridDim.x * (blockDim.x >> 5);
    for (int e = wid; e < E; e += nw) {
        if (etype[e] != rel) continue;
        const int s = esrc[e];
        const int d = edst[e];
        const float inv = 1.0f / fmaxf(cnt[(size_t)rel * N + d], 1.0f);
        const float* __restrict__ xs = xr + (size_t)s * EMB;
        float* __restrict__ hd = hout + (size_t)d * EMB;
        #pragma unroll
        for (int i = 0; i < 4; ++i) {
            const int f = lane + (i << 5);
            __hip_atomic_fetch_add(&hd[f], xs[f] * inv,
                                   __ATOMIC_RELAXED, __HIP_MEMORY_SCOPE_AGENT);
        }
    }
}

__global__ void relu_kernel(float* __restrict__ p, int n) {
    int i = blockIdx.x * blockDim.x + threadIdx.x;
    if (i < n) p[i] = fmaxf(p[i], 0.0f);
}

// combined[b] = concat(h[nest[b]], h[food[b]])
__global__ void gather_kernel(const float* __restrict__ h,
                              const int* __restrict__ nest,
                              const int* __restrict__ food,
                              float* __restrict__ comb) {
    int i = blockIdx.x * blockDim.x + threadIdx.x;
    if (i < BQ * 2 * EMB) {
        int b = i >> 8;
        int j = i & 255;
        int node = (j < EMB) ? nest[b] : food[b];
        comb[i] = h[(size_t)node * EMB + (j & (EMB - 1))];
    }
}

// dir logits + log_softmax + distance scalar, one thread per query row
__global__ void head_kernel(const float* __restrict__ hidden,
                            const float* __restrict__ dirW,
                            const float* __restrict__ dirb,
                            const float* __restrict__ distW,
                            const float* __restrict__ distb,
                            float* __restrict__ out) {
    int b = blockIdx.x * blockDim.x + threadIdx.x;
    if (b >= BQ) return;
    const float* __restrict__ hrow = hidden + (size_t)b * HID;
    float logit[VOCAB];
    #pragma unroll
    for (int v = 0; v < VOCAB; ++v) logit[v] = dirb[v];
    float dist = distb[0];
    for (int k = 0; k < HID; ++k) {
        const float hv = hrow[k];
        #pragma unroll
        for (int v = 0; v < VOCAB; ++v) logit[v] = fmaf(hv, dirW[k * VOCAB + v], logit[v]);
        dist = fmaf(hv, distW[k], dist);
    }
    float m = logit[0];
    #pragma unroll
    for (int v = 1; v < VOCAB; ++v) m = fmaxf(m, logit[v]);
    float s = 0.0f;
    #pragma unroll
    for (int v = 0; v < VOCAB; ++v) s += __expf(logit[v] - m);
    const float lse = m + __logf(s);
    #pragma unroll
    for (int v = 0; v < VOCAB; ++v) out[(size_t)b * VOCAB + v] = logit[v] - lse;
    out[(size_t)BQ * VOCAB + b] = dist;
}

// ------------------------------------------------------------------ driver
extern "C" void kernel_launch(void* const* d_in, const int* in_sizes, int n_in,
                              void* d_out, int out_size, void* d_ws, size_t ws_size,
                              hipStream_t stream) {
    const float* x      = (const float*)d_in[0];
    const int*   eidx   = (const int*)d_in[1];     // [2,E]: src row then dst row
    const int*   etype  = (const int*)d_in[2];
    const int*   nest   = (const int*)d_in[3];
    const int*   food   = (const int*)d_in[4];
    const float* Wrel1  = (const float*)d_in[5];   // [4,128,128]
    const float* Wroot1 = (const float*)d_in[6];
    const float* b1     = (const float*)d_in[7];
    const float* Wrel2  = (const float*)d_in[8];
    const float* Wroot2 = (const float*)d_in[9];
    const float* b2     = (const float*)d_in[10];
    const float* fcW    = (const float*)d_in[11];  // [256,256]
    const float* fcb    = (const float*)d_in[12];
    const float* dirW   = (const float*)d_in[13];  // [256,16]
    const float* dirb   = (const float*)d_in[14];
    const float* distW  = (const float*)d_in[15];  // [256,1]
    const float* distb  = (const float*)d_in[16];
    float* out = (float*)d_out;

    const int* esrc = eidx;
    const int* edst = eidx + NEDGES;

    // workspace layout (floats)
    float* ws = (float*)d_ws;
    size_t off = 0;
    float* cnt    = ws + off; off += (size_t)NREL * NNODES;     // 160000
    float* h1     = ws + off; off += (size_t)NNODES * EMB;      // 5.12M
    float* h2     = ws + off; off += (size_t)NNODES * EMB;
    float* xr     = ws + off; off += (size_t)NNODES * EMB;
    float* comb   = ws + off; off += (size_t)BQ * 2 * EMB;
    float* hidden = ws + off; off += (size_t)BQ * HID;

    const int cntN = NREL * NNODES;
    hipLaunchKernelGGL(zero_kernel, dim3((cntN + 255) / 256), dim3(256), 0, stream, cnt, cntN);
    hipLaunchKernelGGL(count_kernel, dim3((NEDGES + 255) / 256), dim3(256), 0, stream,
                       edst, etype, cnt, NEDGES, NNODES);

    const dim3 gblk(32 * GWAVES);
    const dim3 ggrid_node(NNODES / 16, FDIM / (16 * GWAVES));   // (2500, 2)
    const dim3 sgrid(2048), sblk(256);

    // ---- layer 1: h1 = relu(x@Wroot1 + b1 + sum_r mean_r(x@Wrel1[r]))
    hipLaunchKernelGGL(wmma_gemm_kernel, ggrid_node, gblk, 0, stream,
                       x, Wroot1, b1, h1, NNODES, FDIM, EMB, 0);
    for (int r = 0; r < NREL; ++r) {
        hipLaunchKernelGGL(wmma_gemm_kernel, ggrid_node, gblk, 0, stream,
                           x, Wrel1 + (size_t)r * FDIM * EMB, nullptr, xr,
                           NNODES, FDIM, EMB, 0);
        hipLaunchKernelGGL(scatter_kernel, sgrid, sblk, 0, stream,
                           esrc, edst, etype, cnt, xr, h1, NEDGES, NNODES, r);
    }
    hipLaunchKernelGGL(relu_kernel, dim3((NNODES * EMB + 255) / 256), dim3(256), 0, stream,
                       h1, NNODES * EMB);

    // ---- layer 2: h2 = h1@Wroot2 + b2 + sum_r mean_r(h1@Wrel2[r])
    hipLaunchKernelGGL(wmma_gemm_kernel, ggrid_node, gblk, 0, stream,
                       h1, Wroot2, b2, h2, NNODES, EMB, EMB, 0);
    for (int r = 0; r < NREL; ++r) {
        hipLaunchKernelGGL(wmma_gemm_kernel, ggrid_node, gblk, 0, stream,
                           h1, Wrel2 + (size_t)r * EMB * EMB, nullptr, xr,
                           NNODES, EMB, EMB, 0);
        hipLaunchKernelGGL(scatter_kernel, sgrid, sblk, 0, stream,
                           esrc, edst, etype, cnt, xr, h2, NEDGES, NNODES, r);
    }

    // ---- MLP heads
    hipLaunchKernelGGL(gather_kernel, dim3((BQ * 2 * EMB + 255) / 256), dim3(256), 0, stream,
                       h2, nest, food, comb);
    hipLaunchKernelGGL(wmma_gemm_kernel, dim3(BQ / 16, HID / (16 * GWAVES)), gblk, 0, stream,
                       comb, fcW, fcb, hidden, BQ, 2 * EMB, HID, 1);
    hipLaunchKernelGGL(head_kernel, dim3((BQ + 255) / 256), dim3(256), 0, stream,
                       hidden, dirW, dirb, distW, distb, out);
}